// Graph_Attention_Layer_18459769438461
// MI455X (gfx1250) — compile-verified
//
#include <hip/hip_runtime.h>
#include <hip/hip_bf16.h>

#define N_NODES 8192
#define IN_DIM  256
#define OUT_DIM 64
#define ALPHA   0.2f
#define CB      128   // c-chunk per iteration of the fused pass

typedef __attribute__((ext_vector_type(2))) float v2f;
typedef __attribute__((ext_vector_type(8))) float v8f;

__device__ __forceinline__ float lrelu(float x) { return x > 0.f ? x : ALPHA * x; }

// ---------------------------------------------------------------------------
// Kernel 1: w_h = input @ weights   [8192,256] x [256,64] -> [8192,64]
// One wave per 16-row block; A fragment reused across the 4 n-tiles.
// V_WMMA_F32_16X16X4_F32, K-loop of 64 steps.
// ---------------------------------------------------------------------------
__global__ __launch_bounds__(32) void k_gemm_wh(const float* __restrict__ in,
                                                const float* __restrict__ W,
                                                float* __restrict__ wh) {
  const int r0   = blockIdx.x * 16;
  const int lane = threadIdx.x;
  const int half = lane >> 4;   // 0: K=0,1  1: K=2,3
  const int lr   = lane & 15;

  v8f acc0 = {}, acc1 = {}, acc2 = {}, acc3 = {};
  const float* arow = in + (size_t)(r0 + lr) * IN_DIM + 2 * half;

  for (int k = 0; k < IN_DIM; k += 4) {
    v2f a;
    a.x = arow[k];
    a.y = arow[k + 1];
    const float* wrow = W + (size_t)(k + 2 * half) * OUT_DIM;
    v2f b0, b1, b2, b3;
    b0.x = wrow[lr];           b0.y = wrow[OUT_DIM + lr];
    b1.x = wrow[16 + lr];      b1.y = wrow[OUT_DIM + 16 + lr];
    b2.x = wrow[32 + lr];      b2.y = wrow[OUT_DIM + 32 + lr];
    b3.x = wrow[48 + lr];      b3.y = wrow[OUT_DIM + 48 + lr];
    acc0 = __builtin_amdgcn_wmma_f32_16x16x4_f32(false, a, false, b0, (short)0, acc0, false, false);
    acc1 = __builtin_amdgcn_wmma_f32_16x16x4_f32(false, a, false, b1, (short)0, acc1, false, false);
    acc2 = __builtin_amdgcn_wmma_f32_16x16x4_f32(false, a, false, b2, (short)0, acc2, false, false);
    acc3 = __builtin_amdgcn_wmma_f32_16x16x4_f32(false, a, false, b3, (short)0, acc3, false, false);
  }

#pragma unroll
  for (int i = 0; i < 8; i++) {
    const int row = r0 + i + 8 * half;           // C layout: VGPR i -> M=i / M=8+i
    wh[(size_t)row * OUT_DIM + lr]      = acc0[i];
    wh[(size_t)row * OUT_DIM + 16 + lr] = acc1[i];
    wh[(size_t)row * OUT_DIM + 32 + lr] = acc2[i];
    wh[(size_t)row * OUT_DIM + 48 + lr] = acc3[i];
  }
}

// ---------------------------------------------------------------------------
// Kernel 2: s1[r] = w_h[r,:] . a[:64]   s2[r] = w_h[r,:] . a[64:128]
// One wave per row, shfl_xor reduction (wave32).
// ---------------------------------------------------------------------------
__global__ __launch_bounds__(256) void k_s1s2(const float* __restrict__ wh,
                                              const float* __restrict__ av,
                                              float* __restrict__ s1,
                                              float* __restrict__ s2) {
  const int lane = threadIdx.x & 31;
  const int row  = blockIdx.x * 8 + (threadIdx.x >> 5);
  const float w0 = wh[(size_t)row * OUT_DIM + lane];
  const float w1 = wh[(size_t)row * OUT_DIM + 32 + lane];
  float p1 = w0 * av[lane]      + w1 * av[32 + lane];
  float p2 = w0 * av[64 + lane] + w1 * av[96 + lane];
#pragma unroll
  for (int off = 16; off > 0; off >>= 1) {
    p1 += __shfl_xor(p1, off, 32);
    p2 += __shfl_xor(p2, off, 32);
  }
  if (lane == 0) { s1[row] = p1; s2[row] = p2; }
}

// ---------------------------------------------------------------------------
// Kernel 3: global max of s2 (single block, deterministic tree reduce).
// Used only as a softmax-stability upper bound (leaky-relu is monotone, so
// lrelu(s1[r] + max_c s2[c]) >= every masked score of row r).
// ---------------------------------------------------------------------------
__global__ __launch_bounds__(1024) void k_maxs2(const float* __restrict__ s2,
                                                float* __restrict__ smax) {
  __shared__ float red[1024];
  float m = -3.4e38f;
  for (int i = threadIdx.x; i < N_NODES; i += 1024) m = fmaxf(m, s2[i]);
  red[threadIdx.x] = m;
  __syncthreads();
  for (int off = 512; off > 0; off >>= 1) {
    if ((int)threadIdx.x < off)
      red[threadIdx.x] = fmaxf(red[threadIdx.x], red[threadIdx.x + off]);
    __syncthreads();
  }
  if (threadIdx.x == 0) smax[0] = red[0];
}

// ---------------------------------------------------------------------------
// Kernel 4: fused masked-softmax attention, single pass over adj (256 MB).
// Block = 128 threads (4 waves) owns 16 rows. Software-pipelined:
//   - adj/s2 for chunk i+1 are issued into registers BEFORE the barrier and
//     stay in flight (LOADcnt) across the whole WMMA phase of chunk i.
//   - P tile is double-buffered in LDS (pitch 132 floats -> strided 64-bit
//     A-fragment reads hit all 64 banks) -> ONE barrier per chunk.
//   phase 1: P[16][CB] = adj ? exp(leaky(s1+s2)-mhat) : 0, denom partials
//   phase 2: 4 waves x V_WMMA_F32_16X16X4_F32 k-loop, B = w_h from L2.
// Epilogue: fixed-order denom reduce (deterministic), divide, ELU, store.
// ---------------------------------------------------------------------------
__global__ __launch_bounds__(128) void k_attn(const float* __restrict__ adj,
                                              const float* __restrict__ wh,
                                              const float* __restrict__ s1,
                                              const float* __restrict__ s2,
                                              const float* __restrict__ smax,
                                              float* __restrict__ out) {
  __shared__ float pt[2][16][CB + 4];   // double-buffered P tile
  __shared__ float dred[16][8];
  __shared__ float dfin[16];

  const int tid  = threadIdx.x;
  const int r0   = blockIdx.x * 16;
  const int prow = tid >> 3;         // 0..15 : row within tile (phase 1)
  const int pseg = tid & 7;          // 0..7  : 16-col segment (phase 1)
  const int wave = tid >> 5;         // 0..3  : n-tile (phase 2)
  const int lane = tid & 31;
  const int half = lane >> 4;
  const int lr   = lane & 15;
  const int n0   = wave * 16;

  const float S2M  = smax[0];
  const float s1r  = s1[r0 + prow];
  const float mrow = lrelu(s1r + S2M);   // >= every masked score in this row

  const float* adjrow = adj + (size_t)(r0 + prow) * N_NODES + pseg * 16;
  const float* s2seg  = s2 + pseg * 16;
  float dsum = 0.f;
  v8f acc = {};

  // preload chunk 0 into registers
  float4 areg[4], sreg[4];
#pragma unroll
  for (int q = 0; q < 4; q++) {
    areg[q] = ((const float4*)adjrow)[q];
    sreg[q] = ((const float4*)s2seg)[q];
  }

  int buf = 0;
  for (int cb = 0; cb < N_NODES; cb += CB, buf ^= 1) {
    // ---- phase 1: build P tile for this chunk from the preloaded regs ----
#pragma unroll
    for (int q = 0; q < 4; q++) {
      const float4 av = areg[q];
      const float4 sv = sreg[q];
      float4 pv;
      float e;
      e = lrelu(s1r + sv.x); pv.x = (av.x != 0.f) ? __expf(e - mrow) : 0.f;
      e = lrelu(s1r + sv.y); pv.y = (av.y != 0.f) ? __expf(e - mrow) : 0.f;
      e = lrelu(s1r + sv.z); pv.z = (av.z != 0.f) ? __expf(e - mrow) : 0.f;
      e = lrelu(s1r + sv.w); pv.w = (av.w != 0.f) ? __expf(e - mrow) : 0.f;
      dsum += (pv.x + pv.y) + (pv.z + pv.w);
      *(float4*)&pt[buf][prow][pseg * 16 + q * 4] = pv;
    }

    // ---- issue next chunk's loads; they overlap the WMMA phase below ----
    if (cb + CB < N_NODES) {
      const float4* a4n = (const float4*)(adjrow + cb + CB);
      const float4* s4n = (const float4*)(s2seg + cb + CB);
#pragma unroll
      for (int q = 0; q < 4; q++) {
        areg[q] = a4n[q];
        sreg[q] = s4n[q];
      }
    }

    __syncthreads();   // P tile visible; next-chunk loads remain in flight

    // ---- phase 2: acc += P_tile @ w_h_tile (per-wave 16x16 output) ----
    const float (*ptc)[CB + 4] = pt[buf];
    const float* bbase = wh + (size_t)(cb + 2 * half) * OUT_DIM + n0 + lr;
#pragma unroll 8
    for (int k = 0; k < CB; k += 4) {
      v2f a, b;
      a.x = ptc[lr][k + 2 * half];
      a.y = ptc[lr][k + 2 * half + 1];
      b.x = bbase[(size_t)k * OUT_DIM];
      b.y = bbase[(size_t)(k + 1) * OUT_DIM];
      acc = __builtin_amdgcn_wmma_f32_16x16x4_f32(false, a, false, b, (short)0, acc, false, false);
    }
    // no second barrier: next phase 1 writes pt[buf^1]
  }

  // ---- denominator: fixed-order reduce (deterministic) ----
  __syncthreads();
  dred[prow][pseg] = dsum;
  __syncthreads();
  if (tid < 16) {
    float d = 0.f;
#pragma unroll
    for (int j = 0; j < 8; j++) d += dred[tid][j];
    dfin[tid] = d;
  }
  __syncthreads();

  // ---- normalize + ELU + store ----
#pragma unroll
  for (int i = 0; i < 8; i++) {
    const int row = i + 8 * half;
    float v = acc[i] / dfin[row];
    v = v > 0.f ? v : (__expf(v) - 1.f);   // elu, alpha=1
    out[(size_t)(r0 + row) * OUT_DIM + n0 + lr] = v;
  }
}

// ---------------------------------------------------------------------------
extern "C" void kernel_launch(void* const* d_in, const int* in_sizes, int n_in,
                              void* d_out, int out_size, void* d_ws, size_t ws_size,
                              hipStream_t stream) {
  const float* inp = (const float*)d_in[0];   // [8192,256]
  const float* adj = (const float*)d_in[1];   // [8192,8192]
  const float* W   = (const float*)d_in[2];   // [256,64]
  const float* av  = (const float*)d_in[3];   // [128,1]
  float* out = (float*)d_out;                 // [8192,64]

  float* wh   = (float*)d_ws;                 // 8192*64
  float* s1   = wh + (size_t)N_NODES * OUT_DIM;
  float* s2   = s1 + N_NODES;
  float* smax = s2 + N_NODES;

  k_gemm_wh<<<N_NODES / 16, 32, 0, stream>>>(inp, W, wh);
  k_s1s2  <<<N_NODES / 8, 256, 0, stream>>>(wh, av, s1, s2);
  k_maxs2 <<<1, 1024, 0, stream>>>(s2, smax);
  k_attn  <<<N_NODES / 16, 128, 0, stream>>>(adj, wh, s1, s2, smax, out);
}